// GATLayer_15985868276491
// MI455X (gfx1250) — compile-verified
//
#include <hip/hip_runtime.h>
#include <hip/hip_bf16.h>
#include <math.h>

#define N_NODES 4096
#define N_EDGES 131072
#define C_IN    256
#define C_OUT   256
#define N_HEADS 4
#define C_HEAD  64              // C_OUT / N_HEADS
#define ALPHA   0.2f
#define MASK_WORDS (N_NODES / 32)   // 128 u32 per row

typedef float v2f __attribute__((ext_vector_type(2)));
typedef float v8f __attribute__((ext_vector_type(8)));

// ---------------------------------------------------------------------------
// K1: build boolean adjacency bitmask (dedups duplicate edges, like the
// reference's .at[...].set(True)). mask[i] row: bit j set iff edge i->j.
// ---------------------------------------------------------------------------
__global__ void gat_build_adj(const int* __restrict__ ei,
                              unsigned* __restrict__ mask) {
    int e = blockIdx.x * blockDim.x + threadIdx.x;
    if (e >= N_EDGES) return;
    int i = ei[e];              // edge_index[0][e]
    int j = ei[N_EDGES + e];    // edge_index[1][e]
    atomicOr(&mask[i * MASK_WORDS + (j >> 5)], 1u << (j & 31));
}

// ---------------------------------------------------------------------------
// K2: feats = x @ W^T + bias  via fp32 WMMA 16x16x4.
// One wave (32 threads) computes a 16x16 tile of feats.
//   A fragment (16x4 f32, 2 VGPRs): lanes 0-15 hold K=0,1; lanes 16-31 K=2,3.
//   B fragment (4x16 f32, 2 VGPRs): rows striped across lanes (v0: K=0/K=2,
//   v1: K=1/K=3), N = lane&15.  B[k][o] = W[o][k]  (W is [C_OUT, C_IN]).
// ---------------------------------------------------------------------------
__global__ void gat_proj_wmma(const float* __restrict__ x,
                              const float* __restrict__ W,
                              const float* __restrict__ bias,
                              float* __restrict__ feats) {
    const int lane  = threadIdx.x;           // 0..31, wave32
    const int nbase = blockIdx.x * 16;       // node-row tile
    const int obase = blockIdx.y * 16;       // out-channel tile

    const int m16  = lane & 15;
    const int half = lane >> 4;              // 0: K=0,1 ; 1: K=2,3

    v8f acc = {};
    const float* xrow = x + (nbase + m16) * C_IN;   // A: row = nbase + m16
    const float* wrow = W + (obase + m16) * C_IN;   // B: col o = obase + m16

    for (int k = 0; k < C_IN; k += 4) {
        int kk = k + half * 2;
        v2f av, bv;
        // contiguous pair -> single b64 load each
        float2 ax = *reinterpret_cast<const float2*>(xrow + kk);
        float2 bx = *reinterpret_cast<const float2*>(wrow + kk);
        av.x = ax.x; av.y = ax.y;
        bv.x = bx.x; bv.y = bx.y;
        acc = __builtin_amdgcn_wmma_f32_16x16x4_f32(
            /*neg_a=*/false, av, /*neg_b=*/false, bv,
            /*c_mod=*/(short)0, acc, /*reuse_a=*/false, /*reuse_b=*/false);
    }

    // D layout: VGPR r: lanes 0-15 -> M=r, lanes 16-31 -> M=r+8 ; N = lane&15
    const int col = obase + m16;
    const float bc = bias[col];
    #pragma unroll
    for (int r = 0; r < 8; ++r) {
        int row = nbase + r + half * 8;
        feats[row * C_OUT + col] = acc[r] + bc;
    }
}

// ---------------------------------------------------------------------------
// K3: per-(node, head) attention scores
//   s_src[n,h] = feats[n,h,:] . a[h, :C],  s_dst[n,h] = feats[n,h,:] . a[h, C:]
// ---------------------------------------------------------------------------
__global__ void gat_scores(const float* __restrict__ feats,
                           const float* __restrict__ a,
                           float* __restrict__ s_src,
                           float* __restrict__ s_dst) {
    int t = blockIdx.x * blockDim.x + threadIdx.x;  // n*H + h
    if (t >= N_NODES * N_HEADS) return;
    int n = t >> 2, h = t & 3;
    const float* f  = feats + n * C_OUT + h * C_HEAD;
    const float* as = a + h * (2 * C_HEAD);
    const float* ad = as + C_HEAD;
    float ss = 0.f, sd = 0.f;
    #pragma unroll 8
    for (int c = 0; c < C_HEAD; ++c) {
        float v = f[c];
        ss += v * as[c];
        sd += v * ad[c];
    }
    s_src[t] = ss;
    s_dst[t] = sd;
}

// ---------------------------------------------------------------------------
// K4: per-destination-node softmax over neighbors + weighted aggregation.
// One block (256 threads) per node i. Deterministic: serial prefix-sum
// compaction, fixed-stride partial reductions, fixed tree combine.
// ---------------------------------------------------------------------------
__global__ void gat_aggregate(const unsigned* __restrict__ mask,
                              const float* __restrict__ feats,
                              const float* __restrict__ s_src,
                              const float* __restrict__ s_dst,
                              float* __restrict__ out) {
    __shared__ int   s_nbr[N_NODES];        // 16 KB (worst-case full row)
    __shared__ int   s_scan[MASK_WORDS];    // 512 B
    __shared__ float s_red[256 * 4];        // 4 KB reduction scratch
    __shared__ int   s_degree;

    const int i   = blockIdx.x;
    const int tid = threadIdx.x;
    const unsigned* mrow = mask + i * MASK_WORDS;

    // --- compaction: popcount + serial exclusive scan + ordered emit -------
    if (tid < MASK_WORDS) s_scan[tid] = __popc(mrow[tid]);
    __syncthreads();
    if (tid == 0) {
        int run = 0;
        for (int w = 0; w < MASK_WORDS; ++w) {
            int c = s_scan[w]; s_scan[w] = run; run += c;
        }
        s_degree = run;
    }
    __syncthreads();
    if (tid < MASK_WORDS) {
        unsigned m = mrow[tid];
        int pos = s_scan[tid];
        while (m) {
            int bit = __ffs(m) - 1;
            m &= m - 1;
            s_nbr[pos++] = tid * 32 + bit;
        }
    }
    __syncthreads();
    const int deg = s_degree;

    // --- degenerate row: reference softmaxes uniform 1/N over ALL nodes ----
    if (deg == 0) {
        float accm = 0.f;
        for (int j = 0; j < N_NODES; ++j) accm += feats[j * C_OUT + tid];
        out[i * C_OUT + tid] = accm * (1.0f / N_NODES);
        return;
    }

    float ssrc[N_HEADS];
    #pragma unroll
    for (int h = 0; h < N_HEADS; ++h) ssrc[h] = s_src[i * N_HEADS + h];

    // --- pass 1: per-head max logit ----------------------------------------
    float lmax[N_HEADS];
    #pragma unroll
    for (int h = 0; h < N_HEADS; ++h) lmax[h] = -3.4e38f;
    for (int t = tid; t < deg; t += 256) {
        int j = s_nbr[t];
        #pragma unroll
        for (int h = 0; h < N_HEADS; ++h) {
            float v = ssrc[h] + s_dst[j * N_HEADS + h];
            v = (v > 0.f) ? v : ALPHA * v;
            lmax[h] = fmaxf(lmax[h], v);
        }
    }
    #pragma unroll
    for (int h = 0; h < N_HEADS; ++h) s_red[tid * 4 + h] = lmax[h];
    __syncthreads();
    for (int s = 128; s > 0; s >>= 1) {
        if (tid < s) {
            #pragma unroll
            for (int h = 0; h < N_HEADS; ++h)
                s_red[tid * 4 + h] = fmaxf(s_red[tid * 4 + h], s_red[(tid + s) * 4 + h]);
        }
        __syncthreads();
    }
    float mx[N_HEADS];
    #pragma unroll
    for (int h = 0; h < N_HEADS; ++h) mx[h] = s_red[h];
    __syncthreads();

    // --- pass 2: per-head sum of exp(l - max) ------------------------------
    float lsum[N_HEADS];
    #pragma unroll
    for (int h = 0; h < N_HEADS; ++h) lsum[h] = 0.f;
    for (int t = tid; t < deg; t += 256) {
        int j = s_nbr[t];
        #pragma unroll
        for (int h = 0; h < N_HEADS; ++h) {
            float v = ssrc[h] + s_dst[j * N_HEADS + h];
            v = (v > 0.f) ? v : ALPHA * v;
            lsum[h] += __expf(v - mx[h]);
        }
    }
    #pragma unroll
    for (int h = 0; h < N_HEADS; ++h) s_red[tid * 4 + h] = lsum[h];
    __syncthreads();
    for (int s = 128; s > 0; s >>= 1) {
        if (tid < s) {
            #pragma unroll
            for (int h = 0; h < N_HEADS; ++h)
                s_red[tid * 4 + h] += s_red[(tid + s) * 4 + h];
        }
        __syncthreads();
    }
    const int hme = tid >> 6;                 // head owning channel tid
    const float inv = 1.0f / s_red[hme];
    const float mh  = mx[hme];
    const float sh  = ssrc[hme];

    // --- pass 3: out[i, c] = sum_j p(i,j,h(c)) * feats[j, c] ---------------
    float acc = 0.f;
    for (int t = 0; t < deg; ++t) {
        int j = s_nbr[t];
        float v = sh + s_dst[j * N_HEADS + hme];
        v = (v > 0.f) ? v : ALPHA * v;
        float p = __expf(v - mh) * inv;
        acc += p * feats[j * C_OUT + tid];    // coalesced row gather (L2 hit)
    }
    out[i * C_OUT + tid] = acc;
}

// ---------------------------------------------------------------------------
extern "C" void kernel_launch(void* const* d_in, const int* in_sizes, int n_in,
                              void* d_out, int out_size, void* d_ws, size_t ws_size,
                              hipStream_t stream) {
    const float* x    = (const float*)d_in[0];
    const int*   ei   = (const int*)  d_in[1];   // [2, N_EDGES]
    const float* W    = (const float*)d_in[2];   // [C_OUT, C_IN]
    const float* bias = (const float*)d_in[3];   // [C_OUT]
    const float* a    = (const float*)d_in[4];   // [H, 2*C_HEAD]
    float* out = (float*)d_out;

    char* ws = (char*)d_ws;
    unsigned* mask  = (unsigned*)ws;                                  // 2 MB
    float*    feats = (float*)(ws + (size_t)2 * 1024 * 1024);         // 4 MB
    float*    s_src = (float*)(ws + (size_t)6 * 1024 * 1024);         // 64 KB
    float*    s_dst = s_src + N_NODES * N_HEADS;                      // 64 KB

    hipMemsetAsync(mask, 0, (size_t)N_NODES * MASK_WORDS * sizeof(unsigned), stream);

    gat_build_adj<<<dim3(N_EDGES / 256), dim3(256), 0, stream>>>(ei, mask);

    gat_proj_wmma<<<dim3(N_NODES / 16, C_OUT / 16), dim3(32), 0, stream>>>(
        x, W, bias, feats);

    gat_scores<<<dim3((N_NODES * N_HEADS) / 256), dim3(256), 0, stream>>>(
        feats, a, s_src, s_dst);

    gat_aggregate<<<dim3(N_NODES), dim3(256), 0, stream>>>(
        mask, feats, s_src, s_dst, out);
}